// ConvJac_13408887898252
// MI455X (gfx1250) — compile-verified
//
#include <hip/hip_runtime.h>
#include <stdint.h>

#define B_ 16
#define H_ 512
#define W_ 512
#define N_ (B_ * H_ * W_)

#define TX 64
#define TY 16
// Halo tile: rows by0-1 .. by0+16 (18), cols bx0-4 .. bx0+67 (72 floats = 18x 16B chunks)
#define HR 18
#define HC 72

typedef unsigned int u32x4 __attribute__((ext_vector_type(4)));
typedef unsigned int u32x8 __attribute__((ext_vector_type(8)));

// ---------------------------------------------------------------------------
// Setup: fold inv_diag into everything once.
//   BI[g]      = b/diag
//   C4[g]      = { left/diag, right/diag, low/diag, (i<H-1 ? up/diag : 0) }
// (south value is Dirichlet-0 on the last row, so zeroing UP there removes
//  the only boundary predicate from the sweep's inner loop)
// ---------------------------------------------------------------------------
__global__ __launch_bounds__(256) void jac_setup(const float* __restrict__ K,
                                                 const float* __restrict__ b,
                                                 float* __restrict__ BI,
                                                 float4* __restrict__ C4) {
  int g = blockIdx.x * 256 + threadIdx.x;
  if (g >= N_) return;
  int j = g & (W_ - 1);
  int i = (g >> 9) & (H_ - 1);

  // clamped, unconditional neighbor loads (no exec-mask branching)
  float kc = K[g];
  float kl = K[j > 0 ? g - 1 : g];
  float kr = K[j < W_ - 1 ? g + 1 : g];
  float ku = K[i > 0 ? g - W_ : g];
  float kd = K[i < H_ - 1 ? g + W_ : g];

  float lbdc = 1.0f / kc;
  float left  = (j > 0)      ? 2.0f / (lbdc + 1.0f / kl) : 0.0f;
  float right = (j < W_ - 1) ? 2.0f / (lbdc + 1.0f / kr) : 0.0f;
  float low   = (i > 0)      ? 2.0f / (lbdc + 1.0f / ku) : 0.0f;
  float up    = (i < H_ - 1) ? 2.0f / (lbdc + 1.0f / kd) : 2.0f / lbdc;

  float inv = 1.0f / (left + right + up + low);
  BI[g] = b[g] * inv;
  float4 c;
  c.x = left * inv;
  c.y = right * inv;
  c.z = low * inv;
  c.w = (i < H_ - 1) ? up * inv : 0.0f;
  C4[g] = c;
}

// ---------------------------------------------------------------------------
// One Jacobi sweep. Interior blocks: TDM tensor_load_to_lds (one 2D descriptor
// per wave, 2-3 rows each, TENSORcnt). Edge blocks: clamped b128 async loads
// (ASYNCcnt). Both produce the same 18x72 f32 LDS layout.
// ---------------------------------------------------------------------------
__global__ __launch_bounds__(256) void jac_iter(const float* __restrict__ src,
                                                float* __restrict__ dst,
                                                const float* __restrict__ BI,
                                                const float4* __restrict__ C4) {
  __shared__ __align__(16) float s[HR * HC];

  const int t   = (int)threadIdx.x;
  const int bx0 = (int)blockIdx.x * TX;
  const int by0 = (int)blockIdx.y * TY;
  const unsigned planeBase = (unsigned)blockIdx.z * (unsigned)(H_ * W_);

  const bool interior = (blockIdx.x >= 1) && (blockIdx.x <= (W_ / TX) - 2) &&
                        (blockIdx.y >= 1) && (blockIdx.y <= (H_ / TY) - 2);

  if (interior) {
    // ---- TDM path: tile rows split across the 8 waves ----
    int w = __builtin_amdgcn_readfirstlane(t >> 5);  // scalar wave id
    int rs = (w * HR) >> 3;            // start row for this wave
    int re = ((w + 1) * HR) >> 3;      // end row
    unsigned rows = (unsigned)(re - rs);

    unsigned ldsOff = (unsigned)(uintptr_t)&s[0] + (unsigned)rs * (HC * 4);
    uint64_t gaddr = (uint64_t)(uintptr_t)src +
                     ((uint64_t)planeBase + (uint64_t)(by0 - 1 + rs) * W_ + (uint64_t)(bx0 - 4)) * 4ull;

    // D# group 0: count=1 | lds_addr | global_addr[56:0] | type=2
    u32x4 g0;
    g0.x = 1u;
    g0.y = ldsOff;
    g0.z = (unsigned)gaddr;
    g0.w = (unsigned)(gaddr >> 32) | 0x80000000u;

    // D# group 1: wg_mask=0, data_size=2 (4B), no flags/padding,
    // tensor dims huge (tile provably in-bounds), tile 72 x rows, stride 512
    const unsigned td0 = 0x40000000u, td1 = 0x40000000u;
    u32x8 g1;
    g1.s0 = 0x20000u;                            // data_size=2 at bits [17:16]
    g1.s1 = (td0 & 0xFFFFu) << 16;               // tensor_dim0[15:0] at [31:16]
    g1.s2 = (td0 >> 16) | ((td1 & 0xFFFFu) << 16);
    g1.s3 = (td1 >> 16) | ((unsigned)HC << 16);  // tile_dim0=72 at [127:112]
    g1.s4 = rows;                                // tile_dim1 ; tile_dim2=0
    g1.s5 = (unsigned)W_;                        // tensor_dim0_stride[31:0]
    g1.s6 = 0u;
    g1.s7 = 0u;

    asm volatile("tensor_load_to_lds %0, %1" : : "s"(g0), "s"(g1) : "memory");
    __builtin_amdgcn_s_wait_tensorcnt(0);
  } else {
    // ---- async b128 path with clamped addresses (image edges) ----
    for (int idx = t; idx < HR * (HC / 4); idx += 256) {  // 18 rows * 18 chunks
      int r = idx / (HC / 4);
      int c = idx - r * (HC / 4);
      int gr = by0 - 1 + r;
      gr = gr < 0 ? 0 : (gr > H_ - 1 ? H_ - 1 : gr);
      int gc0 = bx0 - 4 + 4 * c;
      gc0 = gc0 < 0 ? 0 : (gc0 > W_ - 4 ? W_ - 4 : gc0);
      unsigned goff = (planeBase + (unsigned)gr * W_ + (unsigned)gc0) * 4u;
      unsigned loff = (unsigned)(uintptr_t)&s[r * HC + 4 * c];
      asm volatile("global_load_async_to_lds_b128 %0, %1, %2"
                   :
                   : "v"(loff), "v"(goff), "s"(src)
                   : "memory");
    }
    asm volatile("s_wait_asynccnt 0" ::: "memory");
  }
  __syncthreads();

  // ---- compute: 64x16 tile, 4 rows/thread; per output: 1 b128 + 1 b32 load ----
  const int tc  = t & 63;
  const int ty4 = t >> 6;
#pragma unroll
  for (int k = 0; k < 4; ++k) {
    int tr = ty4 + 4 * k;
    int g  = (int)planeBase + (by0 + tr) * W_ + (bx0 + tc);
    float ul = s[(tr + 1) * HC + tc + 3];   // west
    float ur = s[(tr + 1) * HC + tc + 5];   // east
    float un = s[tr * HC + tc + 4];         // north
    float us = s[(tr + 2) * HC + tc + 4];   // south
    float4 c = C4[g];
    float v = BI[g];
    v = fmaf(c.x, ul, v);
    v = fmaf(c.y, ur, v);
    v = fmaf(c.z, un, v);
    v = fmaf(c.w, us, v);
    dst[g] = v;
  }
}

extern "C" void kernel_launch(void* const* d_in, const int* in_sizes, int n_in,
                              void* d_out, int out_size, void* d_ws, size_t ws_size,
                              hipStream_t stream) {
  (void)in_sizes; (void)n_in; (void)out_size; (void)ws_size;
  // inputs: [0]=max_iter (fixed 32), [1]=u, [2]=b, [3]=K
  const float* u = (const float*)d_in[1];
  const float* b = (const float*)d_in[2];
  const float* K = (const float*)d_in[3];
  float* out = (float*)d_out;

  // workspace: ping(N) | BI(N) | C4(4N) = 96 MB (L2-resident)
  float*  ping = (float*)d_ws;
  float*  BI   = ping + N_;
  float4* C4   = (float4*)(BI + N_);

  jac_setup<<<N_ / 256, 256, 0, stream>>>(K, b, BI, C4);

  dim3 grid(W_ / TX, H_ / TY, B_);
  const float* src = u;
  for (int it = 0; it < 32; ++it) {
    float* dst = (it & 1) ? out : ping;  // it=31 (odd) -> final result lands in d_out
    jac_iter<<<grid, 256, 0, stream>>>(src, dst, BI, C4);
    src = dst;
  }
}